// RUL_loss_12635793785120
// MI455X (gfx1250) — compile-verified
//
#include <hip/hip_runtime.h>
#include <math.h>

typedef float v2f __attribute__((ext_vector_type(2)));
typedef float v4f __attribute__((ext_vector_type(4)));
typedef float v8f __attribute__((ext_vector_type(8)));

// ---------------- zero the two workspace accumulators (graph-safe init) ----
__global__ void rul_init_ws(float* __restrict__ ws) {
    if (threadIdx.x < 2) ws[threadIdx.x] = 0.0f;
}

// ---------------- main streaming reduction ---------------------------------
// accS = sum(exp(piecewise) - 1), accQ = sum(diff^2)
__global__ __launch_bounds__(256) void rul_partial(const float* __restrict__ pred,
                                                   const float* __restrict__ tru,
                                                   float* __restrict__ ws,
                                                   long long nvec, long long n) {
    const float kNegScale = -1.0f / 13.0f;   // diff < 0  -> exp(-diff/13)
    const float kPosScale =  0.1f;           // diff >= 0 -> exp(diff/10)

    const long long tid    = (long long)blockIdx.x * blockDim.x + threadIdx.x;
    const long long stride = (long long)gridDim.x * blockDim.x;

    const v4f* pv = (const v4f*)pred;
    const v4f* tv = (const v4f*)tru;

    float accS = 0.0f;
    float accQ = 0.0f;

    // 128-bit non-temporal streaming loads; data is touched exactly once.
    for (long long i = tid; i < nvec; i += stride) {
        v4f a = __builtin_nontemporal_load(pv + i);
        v4f b = __builtin_nontemporal_load(tv + i);
#pragma unroll
        for (int j = 0; j < 4; ++j) {
            float d = a[j] - b[j];
            float s = (d < 0.0f) ? kNegScale : kPosScale;
            float e = __expf(d * s);          // -> v_exp_f32 path
            accS += (e - 1.0f);               // elementwise -1 for accuracy
            accQ  = fmaf(d, d, accQ);
        }
    }
    // scalar tail (n not multiple of 4)
    for (long long i = nvec * 4 + tid; i < n; i += stride) {
        float d = pred[i] - tru[i];
        float s = (d < 0.0f) ? kNegScale : kPosScale;
        accS += (__expf(d * s) - 1.0f);
        accQ  = fmaf(d, d, accQ);
    }

    // ---- wave32 reduction via V_WMMA_F32_16X16X4_F32 ----------------------
    // A: row 0 all ones (A[0][k]=1, k=0..3): per the 16x4 f32 A layout that is
    // lanes {0,16} holding 1.0 in both A VGPRs. B: accumulator replicated into
    // both B VGPRs -> every lane's value is counted exactly twice in
    // D[0][n] = 2*(v(n) + v(n+16)), independent of B's row<->half mapping.
    const int lane = threadIdx.x & 31;
    const float sel = (lane == 0 || lane == 16) ? 1.0f : 0.0f;
    v2f A;  A[0] = sel;  A[1] = sel;
    v8f zero = {};

    v2f Bs; Bs[0] = accS; Bs[1] = accS;
    v8f Ds = __builtin_amdgcn_wmma_f32_16x16x4_f32(
        /*neg_a=*/false, A, /*neg_b=*/false, Bs,
        /*c_mod=*/(short)0, zero, /*reuse_a=*/false, /*reuse_b=*/false);

    v2f Bq; Bq[0] = accQ; Bq[1] = accQ;
    v8f Dq = __builtin_amdgcn_wmma_f32_16x16x4_f32(
        false, A, false, Bq, (short)0, zero, false, false);

    // D vgpr0: lanes 0-15 hold M=0 row (the sums), lanes 16-31 hold M=8 row = 0.
    float rs = Ds[0];
    float rq = Dq[0];
#pragma unroll
    for (int m = 8; m >= 1; m >>= 1) {
        rs += __shfl_xor(rs, m, 32);
        rq += __shfl_xor(rq, m, 32);
    }
    // lane 0 now holds 2 * (wave total)

    __shared__ float sS[8];
    __shared__ float sQ[8];
    const int wave = threadIdx.x >> 5;
    if (lane == 0) { sS[wave] = rs * 0.5f; sQ[wave] = rq * 0.5f; }
    __syncthreads();
    if (threadIdx.x == 0) {
        float bs = 0.0f, bq = 0.0f;
        const int nw = (blockDim.x + 31) >> 5;
        for (int w = 0; w < nw; ++w) { bs += sS[w]; bq += sQ[w]; }
        atomicAdd(ws + 0, bs);   // global_atomic_add_f32
        atomicAdd(ws + 1, bq);
    }
}

// ---------------- finalize: combine with theta -----------------------------
__global__ void rul_final(const float* __restrict__ ws,
                          const float* __restrict__ theta,
                          float* __restrict__ out, long long n) {
    const float th = theta[0];
    const float se = ws[0];          // sum(exp - 1)
    const float sq = ws[1];          // sum(diff^2)
    out[0] = th * se + (1.0f - th) * sqrtf(sq / (float)n);
}

extern "C" void kernel_launch(void* const* d_in, const int* in_sizes, int n_in,
                              void* d_out, int out_size, void* d_ws, size_t ws_size,
                              hipStream_t stream) {
    const float* pred  = (const float*)d_in[0];
    const float* tru   = (const float*)d_in[1];
    const float* theta = (const float*)d_in[2];
    float* ws = (float*)d_ws;

    const long long n    = (long long)in_sizes[0];
    const long long nvec = n >> 2;

    rul_init_ws<<<1, 64, 0, stream>>>(ws);

    const int threads = 256;
    long long want = (nvec + threads - 1) / threads;
    if (want < 1) want = 1;
    const int blocks = (int)(want > 4096 ? 4096 : want);

    rul_partial<<<blocks, threads, 0, stream>>>(pred, tru, ws, nvec, n);
    rul_final<<<1, 1, 0, stream>>>(ws, theta, (float*)d_out, n);
}